// FLANN_19645180412782
// MI455X (gfx1250) — compile-verified
//
#include <hip/hip_runtime.h>
#include <math.h>

// ---------------------------------------------------------------------------
// Problem constants (from reference): B=1024, D=256, L=2, H=8
// ---------------------------------------------------------------------------
#define B_SZ   1024
#define D_SZ   256
#define NHEAD  8
#define EPS_F  1e-8f
#define LN_EPS 1e-5f

typedef float v2f __attribute__((ext_vector_type(2)));
typedef float v8f __attribute__((ext_vector_type(8)));
typedef unsigned int u32x4 __attribute__((ext_vector_type(4)));
typedef int i32x4 __attribute__((ext_vector_type(4)));
typedef int i32x8 __attribute__((ext_vector_type(8)));

// ---------------------------------------------------------------------------
// FLA kernel: one block per batch row. x row + per-head q,v staged in LDS
// (16B aligned so the unrolled j-loop reads merge into ds_load_b128).
// Two-pass softmax over j with raw rcp/exp (transcendental-bound part).
// ---------------------------------------------------------------------------
__global__ __launch_bounds__(D_SZ) void fla_kernel(const float* __restrict__ xin,
                                                   const float* __restrict__ alphas, // [H,3]
                                                   const float* __restrict__ betas,  // [H,3]
                                                   float* __restrict__ xout) {
  __shared__ alignas(16) float xs[D_SZ];
  __shared__ alignas(16) float qs[D_SZ];
  __shared__ alignas(16) float vs[D_SZ];

  const int b = blockIdx.x;
  const int i = threadIdx.x;

  xs[i] = xin[b * D_SZ + i];
  __syncthreads();
  const float xi = xs[i];

  const float scale = 0.0625f; // 1/sqrt(256)
  float acc = 0.0f;

  for (int h = 0; h < NHEAD; ++h) {
    const float a0 = alphas[h * 3 + 0];
    const float a1 = alphas[h * 3 + 1];
    const float a2 = alphas[h * 3 + 2];
    const float b0 = betas[h * 3 + 0];
    const float b1 = betas[h * 3 + 1];
    const float b2 = betas[h * 3 + 2];

    qs[i] = fmaf(a0, xi, b0);
    vs[i] = fmaf(a2, xi, b2);
    __syncthreads();

    const float ki = fmaf(a1, xi, b1);

    // pass 1: row max of s_ij = 1/(|q_j - k_i| + eps)
    float m = -3.402823466e+38f;
#pragma unroll 8
    for (int j = 0; j < D_SZ; ++j) {
      float s = __builtin_amdgcn_rcpf(fabsf(qs[j] - ki) + EPS_F);
      m = fmaxf(m, s);
    }

    // pass 2: exp-sum and weighted sum
    float den = 0.0f, num = 0.0f;
#pragma unroll 8
    for (int j = 0; j < D_SZ; ++j) {
      float s = __builtin_amdgcn_rcpf(fabsf(qs[j] - ki) + EPS_F);
      float e = __expf(s - m);
      den += e;
      num = fmaf(e, vs[j], num);
    }
    acc = fmaf(num * __builtin_amdgcn_rcpf(den), scale, acc);
    __syncthreads(); // protect qs/vs before next head overwrites
  }

  xout[b * D_SZ + i] = xi + acc;
}

// ---------------------------------------------------------------------------
// Fused ReLU + LayerNorm, in place. One block per row, LDS tree reduction.
// ---------------------------------------------------------------------------
__global__ __launch_bounds__(D_SZ) void relu_ln_kernel(float* __restrict__ data,
                                                       const float* __restrict__ g,
                                                       const float* __restrict__ bta) {
  __shared__ float s1[D_SZ];
  __shared__ float s2[D_SZ];

  const int row = blockIdx.x;
  const int i   = threadIdx.x;

  float r = fmaxf(data[row * D_SZ + i], 0.0f);
  s1[i] = r;
  s2[i] = r * r;
  __syncthreads();
#pragma unroll
  for (int off = D_SZ / 2; off > 0; off >>= 1) {
    if (i < off) {
      s1[i] += s1[i + off];
      s2[i] += s2[i + off];
    }
    __syncthreads();
  }
  const float inv_d = 1.0f / (float)D_SZ;
  const float mean  = s1[0] * inv_d;
  const float var   = s2[0] * inv_d - mean * mean;
  const float rstd  = rsqrtf(var + LN_EPS);
  data[row * D_SZ + i] = (r - mean) * rstd * g[i] + bta[i];
}

// ---------------------------------------------------------------------------
// WMMA f32 GEMM: C[1024x256] = A[1024x256] @ W[256x256] + bias.
// 4 waves / block; each wave owns a 16x32 tile (two accumulators sharing the
// A fragment); K stepped by 4 via V_WMMA_F32_16X16X4_F32.
// The 16x256 A tile is staged into LDS by the Tensor Data Mover
// (tensor_load_to_lds), with D# padding producing a 260-float pitch so the
// fragment ds_load_b64 reads are bank-conflict free.
// ---------------------------------------------------------------------------
#define A_PITCH 260   // 256 dwords + 4 pad dwords per row (from TDM pad)

__global__ __launch_bounds__(128) void gemm_wmma_kernel(const float* __restrict__ A,
                                                        const float* __restrict__ W,
                                                        const float* __restrict__ bias,
                                                        float* __restrict__ C) {
  __shared__ alignas(16) float As[16 * A_PITCH];

  const int tid   = threadIdx.x;
  const int wave  = tid >> 5;
  const int lane  = tid & 31;
  const int half  = lane >> 4;   // 0: lanes 0-15, 1: lanes 16-31
  const int lm    = lane & 15;

  const int tileM  = blockIdx.x * 16;
  const int tileN0 = blockIdx.y * 128 + wave * 32;

  // ---- TDM: DMA the 16x256 f32 A tile from global into LDS (wave 0) ----
  if (wave == 0) {
    unsigned long long gaddr = (unsigned long long)(uintptr_t)(A + (size_t)tileM * D_SZ);
    unsigned int lds_base = (unsigned int)(uintptr_t)(&As[0]); // low 32 bits = LDS offset

    u32x4 g0;
    g0[0] = 1u;                                   // count=1 (valid), user mode, no gather
    g0[1] = lds_base;                             // lds_addr (bytes)
    g0[2] = (unsigned int)gaddr;                  // global_addr[31:0]
    g0[3] = ((unsigned int)(gaddr >> 32) & 0x01FFFFFFu) | (2u << 30); // addr[56:32] | type=2

    i32x8 g1;
    g1[0] = (2 << 16)        // data_size = 4 bytes
          | (1 << 20)        // pad_enable
          | (7 << 22)        // pad_interval: every 256 dwords
          | (3 << 25);       // pad_amount: 4 dwords -> 260-float pitch
    g1[1] = (D_SZ & 0xFFFF) << 16;   // tensor_dim0[15:0] = 256 (atomic barrier addr = 0)
    g1[2] = (16 << 16);              // tensor_dim0[31:16]=0 | tensor_dim1[15:0] = 16
    g1[3] = (D_SZ << 16);            // tensor_dim1[31:16]=0 | tile_dim0 = 256
    g1[4] = 16;                      // tile_dim1 = 16, tile_dim2 = 0 (2D)
    g1[5] = D_SZ;                    // tensor_dim0_stride[31:0] = 256 elements
    g1[6] = 0;                       // stride0[47:32] | stride1[15:0]
    g1[7] = 0;                       // stride1[47:16]

    i32x4 gz4 = {0, 0, 0, 0};              // groups 2/3 unused for 2D tile
    i32x8 gz8 = {0, 0, 0, 0, 0, 0, 0, 0};  // trailing descriptor words unused

    __builtin_amdgcn_tensor_load_to_lds(g0, g1, gz4, gz4, gz8, 0);
    __builtin_amdgcn_s_wait_tensorcnt(0);
  }
  __syncthreads();

  v8f c0 = {};   // N = tileN0 .. tileN0+15
  v8f c1 = {};   // N = tileN0+16 .. tileN0+31

#pragma unroll 4
  for (int k0 = 0; k0 < D_SZ; k0 += 4) {
    // A fragment: lane holds M=lm, K = k0 + 2*half + {0,1}
    v2f a;
    a.x = As[lm * A_PITCH + k0 + 2 * half + 0];
    a.y = As[lm * A_PITCH + k0 + 2 * half + 1];

    // B fragments: lane holds N=tileN0+lm (+16), K = k0 + 2*half + {0,1}
    const float* wp = W + (k0 + 2 * half) * D_SZ + tileN0 + lm;
    v2f b0, b1;
    b0.x = wp[0];
    b0.y = wp[D_SZ];
    b1.x = wp[16];
    b1.y = wp[16 + D_SZ];

    c0 = __builtin_amdgcn_wmma_f32_16x16x4_f32(false, a, false, b0,
                                               (short)0, c0, false, false);
    c1 = __builtin_amdgcn_wmma_f32_16x16x4_f32(false, a, false, b1,
                                               (short)0, c1, false, false);
  }

  // C/D layout: VGPR r holds M=r (lanes 0-15) / M=8+r (lanes 16-31), N=lm.
  const int col0 = tileN0 + lm;
  const int col1 = col0 + 16;
  const float bb0 = bias[col0];
  const float bb1 = bias[col1];
#pragma unroll
  for (int r = 0; r < 8; ++r) {
    const int row = tileM + r + 8 * half;
    C[row * D_SZ + col0] = c0[r] + bb0;
    C[row * D_SZ + col1] = c1[r] + bb1;
  }
}

// ---------------------------------------------------------------------------
// Output head: out[b] = dot(x[b,:], out_W[:,0]) + out_b
// ---------------------------------------------------------------------------
__global__ __launch_bounds__(D_SZ) void out_kernel(const float* __restrict__ x,
                                                   const float* __restrict__ oW,
                                                   const float* __restrict__ ob,
                                                   float* __restrict__ out) {
  __shared__ float s1[D_SZ];
  const int b = blockIdx.x;
  const int i = threadIdx.x;
  s1[i] = x[b * D_SZ + i] * oW[i];
  __syncthreads();
#pragma unroll
  for (int off = D_SZ / 2; off > 0; off >>= 1) {
    if (i < off) s1[i] += s1[i + off];
    __syncthreads();
  }
  if (i == 0) out[b] = s1[0] + ob[0];
}

// ---------------------------------------------------------------------------
// Launcher
// ---------------------------------------------------------------------------
extern "C" void kernel_launch(void* const* d_in, const int* in_sizes, int n_in,
                              void* d_out, int out_size, void* d_ws, size_t ws_size,
                              hipStream_t stream) {
  (void)in_sizes; (void)n_in; (void)out_size; (void)ws_size;

  const float* x   = (const float*)d_in[0];
  const float* fa  = (const float*)d_in[1];   // [L,H,3]
  const float* fb  = (const float*)d_in[2];   // [L,H,3]
  const float* flg = (const float*)d_in[3];   // [L,D]
  const float* flb = (const float*)d_in[4];   // [L,D]
  const float* lW  = (const float*)d_in[5];   // [L,D,D]
  const float* lb  = (const float*)d_in[6];   // [L,D]
  const float* llg = (const float*)d_in[7];   // [L,D]
  const float* llb = (const float*)d_in[8];   // [L,D]
  const float* oW  = (const float*)d_in[9];   // [D,1]
  const float* ob  = (const float*)d_in[10];  // [1]
  float*       out = (float*)d_out;

  float* ping = (float*)d_ws;                 // B*D floats
  float* pong = ping + (size_t)B_SZ * D_SZ;   // B*D floats

  const float* cur = x;
  for (int l = 0; l < 2; ++l) {
    fla_kernel<<<B_SZ, D_SZ, 0, stream>>>(cur, fa + l * NHEAD * 3, fb + l * NHEAD * 3, pong);
    relu_ln_kernel<<<B_SZ, D_SZ, 0, stream>>>(pong, flg + l * D_SZ, flb + l * D_SZ);
    gemm_wmma_kernel<<<dim3(B_SZ / 16, D_SZ / 128), 128, 0, stream>>>(
        pong, lW + (size_t)l * D_SZ * D_SZ, lb + l * D_SZ, ping);
    relu_ln_kernel<<<B_SZ, D_SZ, 0, stream>>>(ping, llg + l * D_SZ, llb + l * D_SZ);
    cur = ping;
  }
  out_kernel<<<B_SZ, D_SZ, 0, stream>>>(cur, oW, ob, out);
}